// RotatedROIPool_17162689314997
// MI455X (gfx1250) — compile-verified
//
#include <hip/hip_runtime.h>
#include <math.h>

// Problem constants (from reference)
#define B_  4
#define N_  32
#define C_  256
#define HF  128
#define WF  128
#define S_  64
static constexpr size_t CROPS_ELEMS = (size_t)B_ * N_ * C_ * S_ * S_;  // 134217728

typedef unsigned int u32x4 __attribute__((ext_vector_type(4)));
typedef int          i32x4 __attribute__((ext_vector_type(4)));
typedef int          i32x8 __attribute__((ext_vector_type(8)));

// ---------------------------------------------------------------------------
// Kernel 1: per-ROI affine params (pre-scaled into feature-pixel space) + M_all
//   ix = (cos*w)*X + (-sin*h)*Y + (cx - 0.5)
//   iy = (sin*w)*X + ( cos*h)*Y + (cy - 0.5)
// with X,Y in [-1,1] over the SxS grid; w/h/cx/cy already /FEAT_DOWN (*EXPAND).
// ---------------------------------------------------------------------------
__global__ __launch_bounds__(128) void obb_params_kernel(
    const float* __restrict__ obb, float* __restrict__ params,
    float* __restrict__ mall) {
  int r = threadIdx.x;
  if (r >= B_ * N_) return;
  const float* o = obb + r * 5;
  float cx = o[0] * 0.125f;
  float cy = o[1] * 0.125f;
  float w  = fmaxf(o[2], 1.0f) * (1.25f * 0.125f);
  float h  = fmaxf(o[3], 1.0f) * (1.25f * 0.125f);
  float ang = o[4] * (3.14159265358979323846f / 180.0f);
  // ang is bounded in [-pi, pi]: native v_sin/v_cos are plenty accurate here
  float sn = __sinf(ang);
  float cs = __cosf(ang);

  float* P = params + r * 8;
  P[0] =  cs * w;        // d(ix)/dX
  P[1] = -sn * h;        // d(ix)/dY
  P[2] =  cx - 0.5f;     // ix at X=Y=0
  P[3] =  sn * w;        // d(iy)/dX
  P[4] =  cs * h;        // d(iy)/dY
  P[5] =  cy - 0.5f;     // iy at X=Y=0
  P[6] = 0.0f; P[7] = 0.0f;

  // M_all = [[A, Bv, Cx], [A2, B2, Cy]]
  float A  =  cs * (w * (1.0f / 64.0f));
  float Bv = -sn * (h * (1.0f / 64.0f));
  float A2 =  sn * (w * (1.0f / 64.0f));
  float B2 =  cs * (h * (1.0f / 64.0f));
  float* M = mall + r * 6;
  M[0] = A;  M[1] = Bv; M[2] = cx - 32.0f * (A + Bv);
  M[3] = A2; M[4] = B2; M[5] = cy - 32.0f * (A2 + B2);
}

// ---------------------------------------------------------------------------
// Kernel 2: one workgroup per (batch, channel). TDM-DMA the 128x128 channel
// plane (64 KB) into LDS (one tensor_load_to_lds issued by wave 0), then
// bilinear-gather all 32 ROIs' 64x64 crops from LDS with coalesced stores.
// ---------------------------------------------------------------------------
__global__ __launch_bounds__(256) void roi_crop_kernel(
    const float* __restrict__ feat, const float* __restrict__ params,
    float* __restrict__ crops) {
  __shared__ float plane[HF * WF];  // 64 KB of the 320 KB WGP LDS

  const int bc = blockIdx.x;        // 0..1023
  const int b  = bc >> 8;           // / C_
  const int c  = bc & (C_ - 1);
  const int t  = threadIdx.x;       // 0..255 (8 wave32 waves)

  const float* gplane = feat + ((size_t)bc << 14);  // 16384 floats per (b,c)
  unsigned ldsBase = (unsigned)(size_t)plane;  // low 32 bits of generic ptr = LDS offset

#if __has_builtin(__builtin_amdgcn_tensor_load_to_lds) && \
    __has_builtin(__builtin_amdgcn_s_wait_tensorcnt)
  // ---- Tensor Data Mover: one DMA for the whole 64 KB plane ----
  if (t < 32) {  // wave 0 issues the descriptor; EXEC is ignored by TDM
    unsigned long long ga = (unsigned long long)(size_t)gplane;
    // D# group0: count=1 | lds_addr | global_addr[56:0] | type=2 ("image")
    u32x4 g0;
    g0[0] = 1u;                                     // count=1, user mode
    g0[1] = ldsBase;                                // lds_addr (bytes)
    g0[2] = (unsigned)(ga & 0xFFFFFFFFull);         // global_addr[31:0]
    g0[3] = (unsigned)((ga >> 32) & 0x01FFFFFFull)  // global_addr[56:32]
            | (2u << 30);                           // type=2
    // D# group1: 1-D tensor of 16384 4-byte elements, tile = whole tensor
    i32x8 g1;
    g1[0] = (int)(2u << 16);        // workgroup_mask=0, data_size=2 (4 bytes)
    g1[1] = (int)(16384u << 16);    // tensor_dim0[15:0] in bits[63:48]
    g1[2] = (int)(1u << 16);        // tensor_dim1[15:0]=1 in bits[95:80]
    g1[3] = (int)(16384u << 16);    // tile_dim0=16384 in bits[127:112]
    g1[4] = 0;                      // tile_dim1=0 (unused), tile_dim2=0
    g1[5] = 16384;                  // tensor_dim0_stride[31:0]
    g1[6] = 0;
    g1[7] = 0;
    i32x4 g2 = {0, 0, 0, 0};
    i32x4 g3 = {0, 0, 0, 0};
#if __clang_major__ >= 23
    i32x8 g4 = {0, 0, 0, 0, 0, 0, 0, 0};
    __builtin_amdgcn_tensor_load_to_lds(g0, g1, g2, g3, g4, 0);
#else
    __builtin_amdgcn_tensor_load_to_lds(g0, g1, g2, g3, 0);
#endif
    __builtin_amdgcn_s_wait_tensorcnt(0);
  }
#else
  // ---- fallback: per-lane async copies (16 B per lane per op) ----
#pragma unroll
  for (int j = 0; j < 16; ++j) {
    int e = (j * 256 + t) * 4;                      // float index
    unsigned lds = ldsBase + (unsigned)(e * 4);     // byte offset in LDS
    unsigned long long g = (unsigned long long)(size_t)(gplane + e);
    asm volatile("global_load_async_to_lds_b128 %0, %1, off"
                 :: "v"(lds), "v"(g)
                 : "memory");
  }
#if __has_builtin(__builtin_amdgcn_s_wait_asynccnt)
  __builtin_amdgcn_s_wait_asynccnt(0);
#else
  asm volatile("s_wait_asynccnt 0x0" ::: "memory");
#endif
#endif
  __syncthreads();

  // Thread t owns column x = t&63, base row y0 = t>>6; covers rows y0+4i.
  const int   xp = t & 63;
  const int   yp = t >> 6;                      // 0..3
  const float X  = (float)xp * (2.0f / 64.0f) + (1.0f / 64.0f - 1.0f);
  const float Y0 = (float)yp * (2.0f / 64.0f) + (1.0f / 64.0f - 1.0f);

  for (int n = 0; n < N_; ++n) {
    const float* P = params + (size_t)(b * N_ + n) * 8;
    const float c11 = P[0], c12 = P[1], c13 = P[2];
    const float c21 = P[3], c22 = P[4], c23 = P[5];
    const float ix0 = c11 * X + c12 * Y0 + c13;
    const float iy0 = c21 * X + c22 * Y0 + c23;
    const float dix = c12 * 0.125f;   // Y step per i: 4 rows = 0.125 in [-1,1]
    const float diy = c22 * 0.125f;

    float* outBase = crops + ((((size_t)(b * N_ + n)) * C_ + c) << 12) + t;

#pragma unroll 4
    for (int i = 0; i < 16; ++i) {
      float ix = fmaf((float)i, dix, ix0);
      float iy = fmaf((float)i, diy, iy0);
      float fx = floorf(ix), fy = floorf(iy);
      float wx = ix - fx,    wy = iy - fy;
      int x0 = (int)fx, y0 = (int)fy;
      int x1 = x0 + 1,  y1 = y0 + 1;
      float vx0 = (x0 >= 0 && x0 < WF) ? 1.0f : 0.0f;
      float vx1 = (x1 >= 0 && x1 < WF) ? 1.0f : 0.0f;
      float vy0 = (y0 >= 0 && y0 < HF) ? 1.0f : 0.0f;
      float vy1 = (y1 >= 0 && y1 < HF) ? 1.0f : 0.0f;
      int xc0 = min(max(x0, 0), WF - 1), xc1 = min(max(x1, 0), WF - 1);
      int yc0 = min(max(y0, 0), HF - 1), yc1 = min(max(y1, 0), HF - 1);
      float w00 = (1.0f - wx) * (1.0f - wy) * vx0 * vy0;
      float w01 = wx * (1.0f - wy) * vx1 * vy0;
      float w10 = (1.0f - wx) * wy * vx0 * vy1;
      float w11 = wx * wy * vx1 * vy1;
      float r = plane[yc0 * WF + xc0] * w00 + plane[yc0 * WF + xc1] * w01 +
                plane[yc1 * WF + xc0] * w10 + plane[yc1 * WF + xc1] * w11;
      outBase[i * 256] = r;   // p = i*256+t -> coalesced stores
    }
  }
}

extern "C" void kernel_launch(void* const* d_in, const int* in_sizes, int n_in,
                              void* d_out, int out_size, void* d_ws, size_t ws_size,
                              hipStream_t stream) {
  const float* feat = (const float*)d_in[0];
  const float* obb  = (const float*)d_in[1];
  float* crops  = (float*)d_out;
  float* mall   = crops + CROPS_ELEMS;   // M_all concatenated after crops
  float* params = (float*)d_ws;          // 128 ROIs x 8 floats = 4 KB scratch

  obb_params_kernel<<<1, 128, 0, stream>>>(obb, params, mall);
  roi_crop_kernel<<<B_ * C_, 256, 0, stream>>>(feat, params, crops);
}